// BGCNEncoder_12292196401321
// MI455X (gfx1250) — compile-verified
//
#include <hip/hip_runtime.h>
#include <math.h>

typedef __attribute__((ext_vector_type(2))) float v2f;
typedef __attribute__((ext_vector_type(8))) float v8f;

#define IN_CH  128
#define OUT_CH 64
#define BN_EPS 1e-5f

// ---------------------------------------------------------------------------
// 1) h = x @ W using V_WMMA_F32_16X16X4_F32.
//    Block = 256 threads = 8 waves; each wave computes one 16-row stripe of h
//    (four 16x16 N-tiles, accumulating over K=128 in steps of 4).
//    W is staged in LDS in K-pair-major swizzled form:
//      Wswz[(k>>1)*OUT_CH + c] = float2{ W[k][c], W[k+1][c] }   (k even)
//    so every B fragment is ONE aligned ds_load_b64 that lands directly in a
//    consecutive VGPR pair in WMMA B-layout (no v_mov shuffling).
// ---------------------------------------------------------------------------
__global__ __launch_bounds__(256) void gcn_gemm_wmma(
    const float* __restrict__ x, const float* __restrict__ W,
    float* __restrict__ h, int N) {
  __shared__ float2 Wswz[(IN_CH / 2) * OUT_CH];   // 32 KB

  const int tid = threadIdx.x;
  // Stage + swizzle W: pair index p = k/2, column c.
  for (int i = tid; i < (IN_CH / 2) * OUT_CH; i += 256) {
    int p = i >> 6;          // k pair
    int c = i & 63;          // column
    float2 v;
    v.x = W[(2 * p) * OUT_CH + c];
    v.y = W[(2 * p + 1) * OUT_CH + c];
    Wswz[i] = v;
  }
  __syncthreads();

  const int wave = tid >> 5;
  const int lane = tid & 31;
  const int tile = blockIdx.x * 8 + wave;
  const int ntiles = (N + 15) >> 4;
  if (tile >= ntiles) return;   // wave-uniform: EXEC stays all-ones for WMMA

  const int mrow = tile * 16 + (lane & 15);
  const int lrow = mrow < N ? mrow : N - 1;   // clamp for safe loads
  const int lhi  = lane >> 4;                 // 0 for lanes 0-15, 1 for 16-31
  const int koff = lhi << 1;                  // lanes 0-15 -> K+0,1 ; 16-31 -> K+2,3
  const int ncol = lane & 15;

  v8f acc0 = {}, acc1 = {}, acc2 = {}, acc3 = {};

  const float2* __restrict__ Wl = Wswz;
  for (int kb = 0; kb < IN_CH; kb += 4) {
    // A fragment: x[mrow][kb+koff], x[mrow][kb+koff+1]  (one global_load_b64)
    v2f a = *(const v2f*)(x + (long)lrow * IN_CH + kb + koff);

    // B fragments: pair row pb = kb/2 + lhi, 4 column tiles. One b64 each.
    const int pb = (kb >> 1) + lhi;
    v2f b0 = *(const v2f*)&Wl[pb * OUT_CH + ncol];
    v2f b1 = *(const v2f*)&Wl[pb * OUT_CH + 16 + ncol];
    v2f b2 = *(const v2f*)&Wl[pb * OUT_CH + 32 + ncol];
    v2f b3 = *(const v2f*)&Wl[pb * OUT_CH + 48 + ncol];

    acc0 = __builtin_amdgcn_wmma_f32_16x16x4_f32(false, a, false, b0, (short)0, acc0, false, false);
    acc1 = __builtin_amdgcn_wmma_f32_16x16x4_f32(false, a, false, b1, (short)0, acc1, false, false);
    acc2 = __builtin_amdgcn_wmma_f32_16x16x4_f32(false, a, false, b2, (short)0, acc2, false, false);
    acc3 = __builtin_amdgcn_wmma_f32_16x16x4_f32(false, a, false, b3, (short)0, acc3, false, false);
  }

  // C/D layout: lanes 0-15 vgpr j -> row j ; lanes 16-31 vgpr j -> row j+8.
  const int rbase = tile * 16 + (lhi << 3);
  for (int j = 0; j < 8; ++j) {
    const int rr = rbase + j;
    if (rr < N) {
      float* hr = h + (long)rr * OUT_CH;
      hr[ncol]      = acc0[j];
      hr[16 + ncol] = acc1[j];
      hr[32 + ncol] = acc2[j];
      hr[48 + ncol] = acc3[j];
    }
  }
}

// ---------------------------------------------------------------------------
// 2) Degree / normalization helpers
// ---------------------------------------------------------------------------
__global__ void deg_init(float* __restrict__ deg, int N) {
  int i = blockIdx.x * blockDim.x + threadIdx.x;
  if (i < N) deg[i] = 1.0f;   // self-loop contribution
}

__global__ void deg_count(const long long* __restrict__ ei, float* __restrict__ deg,
                          int E) {
  int e = blockIdx.x * blockDim.x + threadIdx.x;
  if (e < E) {
    int d = (int)ei[(long)E + e];   // dst row
    atomicAdd(&deg[d], 1.0f);
  }
}

__global__ void deg_rsqrt(float* __restrict__ deg, int N) {
  int i = blockIdx.x * blockDim.x + threadIdx.x;
  if (i < N) deg[i] = rsqrtf(deg[i]);   // deg >= 1 always (self-loop)
}

__global__ void zero_sums(float* __restrict__ sums) {
  if (threadIdx.x < 128) sums[threadIdx.x] = 0.0f;
}

// ---------------------------------------------------------------------------
// 3) agg init: d_out[n][c] = h[n][c] * dinv[n]^2   (self-loop message)
// ---------------------------------------------------------------------------
__global__ void agg_init(const float* __restrict__ h, const float* __restrict__ dinv,
                         float* __restrict__ out, int N) {
  int idx = blockIdx.x * blockDim.x + threadIdx.x;   // over N * 16
  if (idx >= N * (OUT_CH / 4)) return;
  int n = idx >> 4;
  float s = dinv[n] * dinv[n];
  float4 hv = ((const float4*)h)[idx];
  float4 o;
  o.x = hv.x * s; o.y = hv.y * s; o.z = hv.z * s; o.w = hv.w * s;
  ((float4*)out)[idx] = o;
}

// ---------------------------------------------------------------------------
// 4) Edge scatter: 16 threads per edge, each handles 4 channels (float4
//    gather of h[src], 4 float atomics). h and out fit in the 192MB L2,
//    so all random traffic stays on-chip.
// ---------------------------------------------------------------------------
__global__ void edge_agg(const long long* __restrict__ ei, const float* __restrict__ h,
                         const float* __restrict__ dinv, float* __restrict__ out,
                         int E) {
  long gtid = (long)blockIdx.x * blockDim.x + threadIdx.x;
  long e = gtid >> 4;
  if (e >= E) return;
  int cq = ((int)gtid & 15) * 4;
  int s = (int)ei[e];
  int d = (int)ei[(long)E + e];
  float norm = dinv[s] * dinv[d];
  float4 hv = *(const float4*)(h + (long)s * OUT_CH + cq);
  float* op = out + (long)d * OUT_CH + cq;
  atomicAdd(op + 0, hv.x * norm);
  atomicAdd(op + 1, hv.y * norm);
  atomicAdd(op + 2, hv.z * norm);
  atomicAdd(op + 3, hv.w * norm);
}

// ---------------------------------------------------------------------------
// 5) a = tanh(agg + bias) written back; per-channel sum & sumsq via atomics.
// ---------------------------------------------------------------------------
__global__ __launch_bounds__(256) void stats_tanh(
    float* __restrict__ out, const float* __restrict__ bias,
    float* __restrict__ sums, int N) {
  int t = blockIdx.x * blockDim.x + threadIdx.x;   // 0 .. 32767
  int c = t & 63;
  int n0 = t >> 6;                                  // 0 .. 511
  const int stride = (128 * 256) >> 6;              // 512
  float b = bias[c];
  float s = 0.f, s2 = 0.f;
  for (int n = n0; n < N; n += stride) {
    float v = tanhf(out[(long)n * OUT_CH + c] + b);
    out[(long)n * OUT_CH + c] = v;
    s += v;
    s2 += v * v;
  }
  atomicAdd(&sums[c], s);
  atomicAdd(&sums[64 + c], s2);
}

// ---------------------------------------------------------------------------
// 6) BatchNorm finalize (biased variance)
// ---------------------------------------------------------------------------
__global__ void bn_finalize(float* __restrict__ out, const float* __restrict__ sums,
                            const float* __restrict__ gamma, const float* __restrict__ beta,
                            int N) {
  long idx = (long)blockIdx.x * blockDim.x + threadIdx.x;
  if (idx >= (long)N * OUT_CH) return;
  int c = (int)(idx & 63);
  float invN = 1.0f / (float)N;
  float mean = sums[c] * invN;
  float var = sums[64 + c] * invN - mean * mean;   // E[a^2] - mean^2
  float scale = gamma[c] * rsqrtf(var + BN_EPS);
  out[idx] = scale * (out[idx] - mean) + beta[c];
}

// ---------------------------------------------------------------------------
extern "C" void kernel_launch(void* const* d_in, const int* in_sizes, int n_in,
                              void* d_out, int out_size, void* d_ws, size_t ws_size,
                              hipStream_t stream) {
  const float*     x     = (const float*)d_in[0];
  const long long* ei    = (const long long*)d_in[1];  // int64 [2, E] flat
  const float*     W     = (const float*)d_in[2];
  const float*     bias  = (const float*)d_in[3];
  const float*     gamma = (const float*)d_in[4];
  const float*     beta  = (const float*)d_in[5];
  float*           out   = (float*)d_out;

  const int N = in_sizes[0] / IN_CH;       // 50000
  const int E = in_sizes[1] / 2;           // 800000

  // Workspace layout
  float* h    = (float*)d_ws;              // N * OUT_CH
  float* dinv = h + (size_t)N * OUT_CH;    // N   (deg, then rsqrt in place)
  float* sums = dinv + N;                  // 128 (sum[64], sumsq[64])

  // 1) GEMM h = x @ W  (WMMA fp32)
  {
    int ntiles = (N + 15) / 16;
    int blocks = (ntiles + 7) / 8;
    gcn_gemm_wmma<<<blocks, 256, 0, stream>>>(x, W, h, N);
  }
  // 2) degree -> dinv
  deg_init<<<(N + 255) / 256, 256, 0, stream>>>(dinv, N);
  deg_count<<<(E + 255) / 256, 256, 0, stream>>>(ei, dinv, E);
  deg_rsqrt<<<(N + 255) / 256, 256, 0, stream>>>(dinv, N);
  // 3) stats accumulators
  zero_sums<<<1, 128, 0, stream>>>(sums);
  // 4) agg init (self-loop term) directly into d_out
  {
    int total = N * (OUT_CH / 4);
    agg_init<<<(total + 255) / 256, 256, 0, stream>>>(h, dinv, out, N);
  }
  // 5) edge scatter-add
  {
    long total = (long)E * 16;
    int blocks = (int)((total + 255) / 256);
    edge_agg<<<blocks, 256, 0, stream>>>(ei, h, dinv, out, E);
  }
  // 6) tanh + per-channel stats
  stats_tanh<<<128, 256, 0, stream>>>(out, bias, sums, N);
  // 7) batchnorm finalize
  {
    long total = (long)N * OUT_CH;
    int blocks = (int)((total + 255) / 256);
    bn_finalize<<<blocks, 256, 0, stream>>>(out, sums, gamma, beta, N);
  }
}